// Proto_82935818486346
// MI455X (gfx1250) — compile-verified
//
#include <hip/hip_runtime.h>

typedef __attribute__((ext_vector_type(2))) float v2f;
typedef __attribute__((ext_vector_type(8))) float v8f;

#define KC   512
#define DIM  64
#define NPOS 2048
#define BATCH 64

// ---------------------------------------------------------------------------
// Kernel 0: c2[k] = ||centers_k||^2, w[k] = softmax(psi)[k]  -> ws[0:512], ws[512:1024]
// ---------------------------------------------------------------------------
__global__ __launch_bounds__(512) void prep_kernel(const float* __restrict__ centers,
                                                   const float* __restrict__ psi,
                                                   float* __restrict__ c2w) {
    __shared__ float red[512];
    const int k = threadIdx.x;

    // squared norm of center k
    const float4* c4 = (const float4*)(centers + (size_t)k * DIM);
    float s = 0.f;
#pragma unroll
    for (int j = 0; j < DIM / 4; ++j) {
        float4 v = c4[j];
        s += v.x * v.x + v.y * v.y + v.z * v.z + v.w * v.w;
    }
    c2w[k] = s;

    // softmax over psi (KC == 512 == blockDim)
    float p = psi[k];
    red[k] = p;
    __syncthreads();
    for (int off = 256; off > 0; off >>= 1) {
        if (k < off) red[k] = fmaxf(red[k], red[k + off]);
        __syncthreads();
    }
    float m = red[0];
    __syncthreads();
    float e = __expf(p - m);
    red[k] = e;
    __syncthreads();
    for (int off = 256; off > 0; off >>= 1) {
        if (k < off) red[k] += red[k + off];
        __syncthreads();
    }
    float denom = red[0];
    c2w[KC + k] = e / denom;
}

// ---------------------------------------------------------------------------
// Kernel 1: zero outputs
// ---------------------------------------------------------------------------
__global__ void zero_kernel(float* __restrict__ out, int n) {
    int i = blockIdx.x * blockDim.x + threadIdx.x;
    if (i < n) out[i] = 0.0f;
}

// ---------------------------------------------------------------------------
// Kernel 2: fused GEMM(distance) + exp + weighted reduce.
// One wave32 per 16-row tile of flattened (B*N).  8 waves / block.
// ---------------------------------------------------------------------------
__global__ __launch_bounds__(256) void score_kernel(const float* __restrict__ F,
                                                    const float* __restrict__ centers,
                                                    const float* __restrict__ c2w,
                                                    const float* __restrict__ beta_p,
                                                    float* __restrict__ out) {
    const float beta = beta_p[0];
    const int lane = threadIdx.x & 31;
    const int wave = threadIdx.x >> 5;
    const int sub  = lane & 15;   // column / row index within 16
    const int half = lane >> 4;   // selects K pair (A/B) or row+8 (C)

    const int tile    = blockIdx.x * 8 + wave;  // 16-row tile id
    const int rowBase = tile * 16;
    const int b       = rowBase >> 11;          // rowBase / NPOS (N = 2048)

    // ---- A fragments: 16 k-steps of the 16x4 fp32 A matrix, kept resident.
    // A layout: lane(sub)=row M, half selects K-pair, vgpr j -> K = 2*half + j.
    v2f a[16];
    const float* Arow = F + (size_t)(rowBase + sub) * DIM + 2 * half;
#pragma unroll
    for (int ks = 0; ks < 16; ++ks)
        a[ks] = *(const v2f*)(Arow + ks * 4);

    // ---- ||f||^2 for my row (lane 'sub' owns row rowBase+sub; duplicated in both halves)
    float f2 = 0.f;
    {
        const float4* r4 = (const float4*)(F + (size_t)(rowBase + sub) * DIM);
#pragma unroll
        for (int j = 0; j < DIM / 4; ++j) {
            float4 v = r4[j];
            f2 += v.x * v.x + v.y * v.y + v.z * v.z + v.w * v.w;
        }
    }
    // Redistribute to C layout: vgpr v holds row (v + 8*half); f2 for row r lives in lane r.
    float f2v[8];
#pragma unroll
    for (int v = 0; v < 8; ++v)
        f2v[v] = __shfl(f2, v + 8 * half, 32);

    const float* w = c2w + KC;
    float s_acc = 0.0f;

#pragma unroll 1
    for (int ct = 0; ct < KC / 16; ++ct) {
        const int colBase = ct * 16;
        const float c2l = c2w[colBase + sub];  // per-lane column values
        const float wl  = w[colBase + sub];

        // B fragments: 4x16 fp32, B[k][n] = centers[n][k] (centers row-major KCxD)
        v2f bfrag[16];
        const float* Brow = centers + (size_t)(colBase + sub) * DIM + 2 * half;
#pragma unroll
        for (int ks = 0; ks < 16; ++ks)
            bfrag[ks] = *(const v2f*)(Brow + ks * 4);

        v8f c = {};
#pragma unroll
        for (int ks = 0; ks < 16; ++ks)
            c = __builtin_amdgcn_wmma_f32_16x16x4_f32(
                    /*neg_a=*/false, a[ks],
                    /*neg_b=*/false, bfrag[ks],
                    /*c_mod=*/(short)0, c,
                    /*reuse_a=*/false, /*reuse_b=*/false);

        // epilogue: dist^2 -> exp -> weighted accumulate (8 elems per lane)
#pragma unroll
        for (int v = 0; v < 8; ++v) {
            float d2 = fmaxf(f2v[v] + c2l - 2.0f * c[v], 0.0f);
            s_acc += wl * __expf(-beta * d2);
        }
    }

    // wave-level reduction over all 32 lanes (sums the whole 16x512 tile block)
#pragma unroll
    for (int off = 16; off > 0; off >>= 1)
        s_acc += __shfl_xor(s_acc, off, 32);

    if (lane == 0)
        atomicAdd(out + b, s_acc * (1.0f / (float)NPOS));
}

// ---------------------------------------------------------------------------
extern "C" void kernel_launch(void* const* d_in, const int* in_sizes, int n_in,
                              void* d_out, int out_size, void* d_ws, size_t ws_size,
                              hipStream_t stream) {
    const float* F_active = (const float*)d_in[0];
    const float* centers  = (const float*)d_in[1];
    const float* psi      = (const float*)d_in[2];
    const float* beta     = (const float*)d_in[3];
    float* out = (float*)d_out;
    float* c2w = (float*)d_ws;  // [KC] c2 followed by [KC] softmax weights

    prep_kernel<<<1, 512, 0, stream>>>(centers, psi, c2w);
    zero_kernel<<<1, 64, 0, stream>>>(out, out_size);

    const int row_tiles = (BATCH * NPOS) / 16;  // 8192
    score_kernel<<<row_tiles / 8, 256, 0, stream>>>(F_active, centers, c2w, beta, out);
}